// RippleNet_57483842289961
// MI455X (gfx1250) — compile-verified
//
#include <hip/hip_runtime.h>

#define BATCH   4096
#define NMEM    32
#define DIM     32
#define NHOP    2
#define NREL    64
#define TB      32          // batch tile per workgroup
#define NTHREADS 256
#define NWAVES  8
#define USTRIDE 33          // padded row stride for U/item/x/W tiles

typedef __attribute__((ext_vector_type(2))) float v2f;
typedef __attribute__((ext_vector_type(8))) float v8f;

__device__ __forceinline__ float wred_sum(float v) {
#pragma unroll
  for (int off = 16; off; off >>= 1) v += __shfl_xor(v, off, 32);
  return v;
}
__device__ __forceinline__ float wred_max(float v) {
#pragma unroll
  for (int off = 16; off; off >>= 1) v = fmaxf(v, __shfl_xor(v, off, 32));
  return v;
}

__global__ __launch_bounds__(NTHREADS, 1)
void ripplenet_fused(const int* __restrict__ items,
                     const int* __restrict__ mem_h,
                     const int* __restrict__ mem_r,
                     const int* __restrict__ mem_t,
                     const int* __restrict__ users,
                     const float* __restrict__ entity,
                     const float* __restrict__ relation,
                     const float* __restrict__ user_tab,
                     const float* __restrict__ W,
                     float* __restrict__ out)
{
  // U[r][b][e] = (item @ R_r)[b][e] for every relation; 64*32 rows, stride 33
  __shared__ float U_lds[NREL * TB * USTRIDE];          // ~270 KB of the 320 KB WGP LDS
  __shared__ float Wlds[DIM * USTRIDE];
  __shared__ float item_lds[TB * USTRIDE];
  __shared__ float x_lds[TB * USTRIDE];
  __shared__ float osum_lds[TB * DIM];
  __shared__ float logits_lds[TB * NMEM];
  __shared__ float probs_lds[TB * NMEM];
  __shared__ int   tidx_lds[TB * NMEM];

  const int tid  = threadIdx.x;
  const int wid  = tid >> 5;
  const int lane = tid & 31;
  const int b0   = blockIdx.x * TB;

  // ---- stage: W (padded), item rows; warm the 256KB relation table into cache ----
  {
    for (int i = tid; i < (NREL * DIM * DIM) / 32; i += NTHREADS)
      __builtin_prefetch(relation + i * 32, 0, 3);            // global_prefetch_b8 per 128B line
    for (int i = tid; i < DIM * DIM; i += NTHREADS)
      Wlds[(i >> 5) * USTRIDE + (i & 31)] = W[i];
    for (int i = tid; i < TB * DIM; i += NTHREADS) {
      const int tb = i >> 5, d = i & 31;
      item_lds[tb * USTRIDE + d] = entity[(size_t)items[b0 + tb] * DIM + d];
      osum_lds[i] = 0.0f;
    }
  }
  __syncthreads();

  // WMMA lane decomposition (ISA 7.12.2: 32-bit A 16x4, C/D 16x16 layouts)
  const int rl     = lane & 15;
  const int kk     = (lane < 16) ? 0 : 2;
  const int rowoff = (lane < 16) ? 0 : 8;

  for (int hop = 0; hop < NHOP; ++hop) {
    const int hbase = hop * BATCH * NMEM;

    // ---- U_r = ItemTile @ R_r for all 64 relations: hot FLOPs on WMMA ----
    // Wave-task = (relation, 16-col N tile); each task computes BOTH 16-row M
    // tiles from ONE B-fragment load -> half the VMEM issue per WMMA vs tiling
    // over (M,N), and 16-deep back-to-back WMMA bursts for the XDL pipe.
    {
      // A fragments (item tile) are relation-invariant for this phase: preload all
      // 8 k-steps for both M tiles into VGPRs once.
      v2f afrag0[8], afrag1[8];
#pragma unroll
      for (int k8 = 0; k8 < 8; ++k8) {
        const int k = 4 * k8 + kk;
        afrag0[k8].x = item_lds[rl * USTRIDE + k];
        afrag0[k8].y = item_lds[rl * USTRIDE + k + 1];
        afrag1[k8].x = item_lds[(16 + rl) * USTRIDE + k];
        afrag1[k8].y = item_lds[(16 + rl) * USTRIDE + k + 1];
      }
      for (int t = wid; t < NREL * 2; t += NWAVES) {
        const int r  = t >> 1;
        const int ni = (t & 1) * 16;
        const float* __restrict__ Rg = relation + r * (DIM * DIM) + ni + rl;
        v2f bfrag[8];
#pragma unroll
        for (int k8 = 0; k8 < 8; ++k8) {          // B[k][e] = R_r[k][e], L2-hot clause
          bfrag[k8].x = Rg[(4 * k8 + kk) * DIM];
          bfrag[k8].y = Rg[(4 * k8 + kk + 1) * DIM];
        }
        v8f acc0 = {}, acc1 = {};
#pragma unroll
        for (int k8 = 0; k8 < 8; ++k8) {
          acc0 = __builtin_amdgcn_wmma_f32_16x16x4_f32(
                     false, afrag0[k8], false, bfrag[k8], (short)0, acc0, false, false);
          acc1 = __builtin_amdgcn_wmma_f32_16x16x4_f32(
                     false, afrag1[k8], false, bfrag[k8], (short)0, acc1, false, false);
        }
#pragma unroll
        for (int j = 0; j < 8; ++j) {
          U_lds[(r * TB + rowoff + j) * USTRIDE + ni + rl]      = acc0[j];
          U_lds[(r * TB + 16 + rowoff + j) * USTRIDE + ni + rl] = acc1[j];
        }
      }
    }
    __syncthreads();

    // ---- logits: logit[b,m] = <U[r(b,m)][b,:], h[b,m,:]>, one wave per slot ----
    for (int s = wid; s < TB * NMEM; s += NWAVES) {
      const int tb = s >> 5, m = s & 31;
      const int mofs = hbase + (b0 + tb) * NMEM + m;
      const int hIdx = mem_h[mofs];
      const int rIdx = mem_r[mofs];
      const float he = entity[(size_t)hIdx * DIM + lane];          // lane = e, 128B coalesced
      const float ue = U_lds[(rIdx * TB + tb) * USTRIDE + lane];   // conflict-free
      const float logit = wred_sum(ue * he);
      if (lane == 0) logits_lds[s] = logit;
    }
    __syncthreads();

    // ---- softmax over memories + weighted memory sum, one wave per b ----
    for (int tb = wid; tb < TB; tb += NWAVES) {
      const int mofs = hbase + (b0 + tb) * NMEM + lane;            // lane = m
      const int tIdx = mem_t[mofs];
      __builtin_prefetch(&entity[(size_t)tIdx * DIM], 0, 0);       // hide t-row gather
      float lg = logits_lds[tb * NMEM + lane];
      const float mx = wred_max(lg);
      const float ex = __expf(lg - mx);
      const float sm = wred_sum(ex);
      probs_lds[tb * NMEM + lane] = ex / sm;
      tidx_lds[tb * NMEM + lane]  = tIdx;

      float od = 0.0f;                                             // lane = d now
#pragma unroll 8
      for (int m = 0; m < NMEM; ++m)
        od = fmaf(probs_lds[tb * NMEM + m],
                  entity[(size_t)tidx_lds[tb * NMEM + m] * DIM + lane], od);
      osum_lds[tb * DIM + lane] += od;
      x_lds[tb * USTRIDE + lane] = item_lds[tb * USTRIDE + lane] + od;
    }
    __syncthreads();

    // ---- item = (item + o) @ W^T via WMMA: 4 waves own the 4 16x16 D tiles ----
    if (wid < 4) {
      const int mi = (wid >> 1) * 16;
      const int ni = (wid & 1) * 16;
      v8f acc = {};
#pragma unroll
      for (int k = 0; k < DIM; k += 4) {
        v2f a, b;
        a.x = x_lds[(mi + rl) * USTRIDE + k + kk];
        a.y = x_lds[(mi + rl) * USTRIDE + k + kk + 1];
        b.x = Wlds[(ni + rl) * USTRIDE + k + kk];    // B[k][n] = W[n][k]
        b.y = Wlds[(ni + rl) * USTRIDE + k + kk + 1];
        acc = __builtin_amdgcn_wmma_f32_16x16x4_f32(
                  false, a, false, b, (short)0, acc, false, false);
      }
#pragma unroll
      for (int j = 0; j < 8; ++j)
        item_lds[(mi + rowoff + j) * USTRIDE + ni + rl] = acc[j];
    }
    __syncthreads();
  }

  // ---- y = o0 + o1 + user ; score = sigmoid(<item, y>) ----
  for (int tb = wid; tb < TB; tb += NWAVES) {
    const int b = b0 + tb;
    const float u  = user_tab[(size_t)users[b] * DIM + lane];
    const float yv = osum_lds[tb * DIM + lane] + u;
    const float dotv = wred_sum(item_lds[tb * USTRIDE + lane] * yv);
    if (lane == 0) out[b] = 1.0f / (1.0f + __expf(-dotv));
  }
}

extern "C" void kernel_launch(void* const* d_in, const int* in_sizes, int n_in,
                              void* d_out, int out_size, void* d_ws, size_t ws_size,
                              hipStream_t stream) {
  (void)in_sizes; (void)n_in; (void)d_ws; (void)ws_size; (void)out_size;
  const int*   items    = (const int*)d_in[0];
  /* d_in[1] = labels (unused by forward scores) */
  const int*   mem_h    = (const int*)d_in[2];
  const int*   mem_r    = (const int*)d_in[3];
  const int*   mem_t    = (const int*)d_in[4];
  const int*   users    = (const int*)d_in[5];
  const float* entity   = (const float*)d_in[6];
  const float* relation = (const float*)d_in[7];
  const float* user_tab = (const float*)d_in[8];
  const float* W        = (const float*)d_in[9];
  float* out = (float*)d_out;

  dim3 grid(BATCH / TB);
  ripplenet_fused<<<grid, NTHREADS, 0, stream>>>(items, mem_h, mem_r, mem_t, users,
                                                 entity, relation, user_tab, W, out);
}